// CNN_Nested_35948876268300
// MI455X (gfx1250) — compile-verified
//
#include <hip/hip_runtime.h>

typedef __attribute__((ext_vector_type(2))) float v2f;
typedef __attribute__((ext_vector_type(8))) float v8f;

// ---- problem constants ----
// B=4, N=256, H=768, D=200, NH=5, HD=40, OUT=12, SZ=25, N_POS=30, hsz=427
// workspace layout (float offsets)
static constexpr int OFF_HEAD = 0;         // 1024*200
static constexpr int OFF_TAIL = 204800;    // 1024*200
static constexpr int OFF_DWH  = 409600;    // 12*201
static constexpr int OFF_DWT  = 412012;    // 12*201
static constexpr int OFF_CS   = 414424;    // 12*30
static constexpr int OFF_UEF  = 414784;    // 12*5*40*40 (stored [o][h][y][x])
static constexpr int OFF_Q    = 510784;    // 48*256*200
static constexpr int OFF_A    = 2968384;   // 48*256
static constexpr int OFF_B    = 2980672;   // 48*256

__device__ __forceinline__ v8f wmma4(v2f a, v2f b, v8f c) {
  // D = A(16x4,f32) * B(4x16,f32) + C(16x16,f32)
  return __builtin_amdgcn_wmma_f32_16x16x4_f32(false, a, false, b, (short)0, c,
                                               false, false);
}

__device__ __forceinline__ float lrelu(float x) {
  return x >= 0.0f ? x : 0.01f * x;
}

// ---------- kernel 1: head/tail = leaky_relu(word_reps @ w.T + b) ----------
// M=1024 (B*N), K=768, N=200 (13 tiles of 16, last partial).
// Out-of-range N columns compute garbage (clamped row pointer) but are never stored.
__global__ void ht_gemm(const float* __restrict__ word,
                        const float* __restrict__ head_w, const float* __restrict__ head_b,
                        const float* __restrict__ tail_w, const float* __restrict__ tail_b,
                        float* __restrict__ ws) {
  const int lane = threadIdx.x & 31;
  const int wave = threadIdx.x >> 5;
  const int mt   = blockIdx.x * 8 + wave;   // 0..63
  const int n0   = blockIdx.y * 16;         // 0..192
  const int sel  = blockIdx.z;              // 0=head, 1=tail
  const float* wgt  = sel ? tail_w : head_w;
  const float* bias = sel ? tail_b : head_b;
  float* dst = ws + (sel ? OFF_TAIL : OFF_HEAD);

  const int m0    = mt * 16;
  const int lrow  = lane & 15;
  const int khalf = (lane >> 4) * 2;
  const float* arow = word + (size_t)(m0 + lrow) * 768;
  const int nb = n0 + lrow;
  const int nclamp = nb < 200 ? nb : 199;          // clamp: no branch in hot loop
  const float* brow = wgt + (size_t)nclamp * 768;

  v8f c = {0.f,0.f,0.f,0.f,0.f,0.f,0.f,0.f};
#pragma unroll 4
  for (int kk = 0; kk < 768; kk += 4) {
    const int kb = kk + khalf;
    v2f a = *(const v2f*)(arow + kb);
    v2f b = *(const v2f*)(brow + kb);
    c = wmma4(a, b, c);
  }
  if (nb < 200) {
    const float bn = bias[nb];
#pragma unroll
    for (int i = 0; i < 8; ++i) {
      const int m = m0 + i + 8 * (lane >> 4);
      dst[(size_t)m * 200 + nb] = lrelu(c[i] + bn);
    }
  }
}

// ---------- kernel 2: fold down_w into Wh/Wt/Ws/U_mh ----------
__global__ void precompute(const float* __restrict__ W, const float* __restrict__ down_w,
                           const float* __restrict__ down_b, const float* __restrict__ size_emb,
                           const float* __restrict__ U_mh, float* __restrict__ ws) {
  const int idx = blockIdx.x * blockDim.x + threadIdx.x;
  if (idx < 2412) {                       // dWh[o][j], j<201
    const int o = idx / 201, j = idx % 201;
    float s = 0.f;
    for (int k = 0; k < 200; ++k) s += down_w[o * 200 + k] * W[k * 427 + j];
    ws[OFF_DWH + idx] = s;
  } else if (idx < 4824) {                // dWt[o][j]
    const int t = idx - 2412;
    const int o = t / 201, j = t % 201;
    float s = 0.f;
    for (int k = 0; k < 200; ++k) s += down_w[o * 200 + k] * W[k * 427 + 201 + j];
    ws[OFF_DWT + t] = s;
  } else if (idx < 5184) {                // Cs[o][s] (+ down_b folded in)
    const int t = idx - 4824;
    const int o = t / 30, sid = t % 30;
    float acc = down_b[o];
    for (int j = 0; j < 25; ++j) {
      float dws = 0.f;
      for (int k = 0; k < 200; ++k) dws += down_w[o * 200 + k] * W[k * 427 + 402 + j];
      acc += dws * size_emb[sid * 25 + j];
    }
    ws[OFF_CS + t] = acc;
  } else if (idx < 101184) {              // UeffT[o][h][y][x] = sum_d dw[o,h*40+d]*U[h,d,x,y]
    const int u = idx - 5184;
    const int o = u / 8000, r = u % 8000;
    const int h = r / 1600, r2 = r % 1600;
    const int y = r2 / 40, x = r2 % 40;
    float s = 0.f;
    for (int d = 0; d < 40; ++d)
      s += down_w[o * 200 + h * 40 + d] * U_mh[((h * 40 + d) * 40 + x) * 40 + y];
    ws[OFF_UEF + ((o * 5 + h) * 40 + y) * 40 + x] = s;
  }
}

// ---------- kernel 3: A[b,o,m] = h_aug . dWh[o] ; Bt[b,o,n] = t_aug . dWt[o] ----------
__global__ void ab_vectors(float* __restrict__ ws) {
  const int idx = blockIdx.x * blockDim.x + threadIdx.x;
  if (idx < 12288) {
    const int bo = idx / 256, m = idx % 256;
    const int b = bo / 12, o = bo % 12;
    const float* hr = ws + OFF_HEAD + (size_t)(b * 256 + m) * 200;
    const float* w  = ws + OFF_DWH + o * 201;
    float s = w[200];                     // "ones" column
    for (int j = 0; j < 200; ++j) s += hr[j] * w[j];
    ws[OFF_A + idx] = s;
  } else if (idx < 24576) {
    const int t = idx - 12288;
    const int bo = t / 256, n = t % 256;
    const int b = bo / 12, o = bo % 12;
    const float* tr = ws + OFF_TAIL + (size_t)(b * 256 + n) * 200;
    const float* w  = ws + OFF_DWT + o * 201;
    float s = w[200];
    for (int j = 0; j < 200; ++j) s += tr[j] * w[j];
    ws[OFF_B + t] = s;
  }
}

// ---------- kernel 4: Q[b,o,m,h*40+y] = head[b,m,h*40+:] @ UeffT[o,h].T ----------
// batched (b,o,h)=240, M=256 (16 tiles), N=40 (3 tiles, last partial), K=40.
// Out-of-range N columns: clamped pointer, garbage, never stored.
__global__ void q_gemm(float* __restrict__ ws) {
  const int lane  = threadIdx.x & 31;
  const int wave  = threadIdx.x >> 5;        // block of 128 -> 4 waves
  const int batch = blockIdx.y;              // (b*12+o)*5 + h
  const int tile  = blockIdx.x * 4 + wave;   // 0..47
  const int b = batch / 60;
  const int o = (batch % 60) / 5;
  const int h = batch % 5;
  const int mt = tile / 3, nt = tile % 3;
  const int m0 = mt * 16, n0 = nt * 16;

  const float* head  = ws + OFF_HEAD;
  const float* ueffT = ws + OFF_UEF + (size_t)(o * 5 + h) * 1600;
  float* Q = ws + OFF_Q + (size_t)(b * 12 + o) * 256 * 200;

  const int lrow  = lane & 15;
  const int khalf = (lane >> 4) * 2;
  const float* arow = head + (size_t)(b * 256 + m0 + lrow) * 200 + h * 40;
  const int nb = n0 + lrow;
  const int nclamp = nb < 40 ? nb : 39;
  const float* brow = ueffT + nclamp * 40;

  v8f c = {0.f,0.f,0.f,0.f,0.f,0.f,0.f,0.f};
#pragma unroll
  for (int kk = 0; kk < 40; kk += 4) {
    const int kb = kk + khalf;
    v2f a  = *(const v2f*)(arow + kb);
    v2f bf = *(const v2f*)(brow + kb);
    c = wmma4(a, bf, c);
  }
  if (nb < 40) {
#pragma unroll
    for (int i = 0; i < 8; ++i) {
      const int m = m0 + i + 8 * (lane >> 4);
      Q[(size_t)m * 200 + h * 40 + nb] = c[i];
    }
  }
}

// ---------- kernel 5: out[b,o,m,n] = Q . tail + A[m] + Bt[n] + Cs[clamp(n-m)] ----------
// 48 batches, M=N=256, K=200; each wave does a 32x32 block (4 accumulators,
// 2 A-fragments x 2 B-fragments reused).
__global__ void out_gemm(const float* __restrict__ ws, float* __restrict__ out) {
  const int lane = threadIdx.x & 31;
  const int wave = threadIdx.x >> 5;         // 8 waves / block
  const int bo = blockIdx.z;                 // 0..47
  const int b = bo / 12, o = bo % 12;
  const int tile = blockIdx.x * 8 + wave;    // 0..63
  const int mt = tile >> 3, nt = tile & 7;
  const int m0 = mt * 32, n0 = nt * 32;

  const float* Q    = ws + OFF_Q + (size_t)bo * 256 * 200;
  const float* V    = ws + OFF_TAIL + (size_t)b * 256 * 200;
  const float* Aarr = ws + OFF_A + bo * 256;
  const float* Barr = ws + OFF_B + bo * 256;
  const float* Cs   = ws + OFF_CS + o * 30;

  const int lrow  = lane & 15;
  const int khalf = (lane >> 4) * 2;
  const float* a0p = Q + (size_t)(m0 + lrow) * 200;
  const float* a1p = Q + (size_t)(m0 + 16 + lrow) * 200;
  const float* b0p = V + (size_t)(n0 + lrow) * 200;
  const float* b1p = V + (size_t)(n0 + 16 + lrow) * 200;

  v8f c00 = {0.f,0.f,0.f,0.f,0.f,0.f,0.f,0.f};
  v8f c01 = c00, c10 = c00, c11 = c00;
#pragma unroll 5
  for (int kk = 0; kk < 200; kk += 4) {
    const int kb = kk + khalf;
    v2f a0 = *(const v2f*)(a0p + kb);
    v2f a1 = *(const v2f*)(a1p + kb);
    v2f b0 = *(const v2f*)(b0p + kb);
    v2f b1 = *(const v2f*)(b1p + kb);
    c00 = wmma4(a0, b0, c00);
    c01 = wmma4(a0, b1, c01);
    c10 = wmma4(a1, b0, c10);
    c11 = wmma4(a1, b1, c11);
  }

  float* obase = out + (size_t)(b * 12 + o) * 256 * 256;
  const int mhi = 8 * (lane >> 4);
  auto store16 = [&](const v8f& c, int ms, int ns) {
    const int n = n0 + ns + lrow;
    const float bn = Barr[n];
#pragma unroll
    for (int i = 0; i < 8; ++i) {
      const int m = m0 + ms + i + mhi;
      int s = n - m;
      s = s < -15 ? -15 : (s > 14 ? 14 : s);
      obase[(size_t)m * 256 + n] = c[i] + Aarr[m] + bn + Cs[s + 15];
    }
  };
  store16(c00, 0, 0);
  store16(c01, 0, 16);
  store16(c10, 16, 0);
  store16(c11, 16, 16);
}

extern "C" void kernel_launch(void* const* d_in, const int* in_sizes, int n_in,
                              void* d_out, int out_size, void* d_ws, size_t ws_size,
                              hipStream_t stream) {
  const float* word     = (const float*)d_in[0];
  // d_in[1] cls_embeding, d_in[2] pieces_index, d_in[3] loss_mask: unused by reference
  const float* head_w   = (const float*)d_in[4];
  const float* head_b   = (const float*)d_in[5];
  const float* tail_w   = (const float*)d_in[6];
  const float* tail_b   = (const float*)d_in[7];
  const float* U_mh     = (const float*)d_in[8];
  const float* size_emb = (const float*)d_in[9];
  const float* W        = (const float*)d_in[10];
  const float* down_w   = (const float*)d_in[11];
  const float* down_b   = (const float*)d_in[12];
  float* ws  = (float*)d_ws;
  float* out = (float*)d_out;

  dim3 htg(8, 13, 2);                               // M-tiles/8, N-tiles, head/tail
  ht_gemm<<<htg, 256, 0, stream>>>(word, head_w, head_b, tail_w, tail_b, ws);
  precompute<<<(101184 + 255) / 256, 256, 0, stream>>>(W, down_w, down_b, size_emb, U_mh, ws);
  ab_vectors<<<24576 / 256, 256, 0, stream>>>(ws);
  dim3 qg(12, 240);                                 // tiles/4, (b,o,h)
  q_gemm<<<qg, 128, 0, stream>>>(ws);
  dim3 og(8, 1, 48);                                // 8 waves * 8 blocks = 64 tiles, (b,o)
  out_gemm<<<og, 256, 0, stream>>>(ws, out);
}